// GE2ELoss_67010079752773
// MI455X (gfx1250) — compile-verified
//
#include <hip/hip_runtime.h>
#include <math.h>

typedef __attribute__((ext_vector_type(2))) float v2f;
typedef __attribute__((ext_vector_type(8))) float v8f;

#define B_   64
#define N_   64
#define M_   10
#define D_   512
#define NM_  (N_ * M_)       // 640
#define EPSF 1e-12f

// ---------------------------------------------------------------------------
// Kernel 1: per-speaker normalized centroid c[b,n,:] = l2norm(mean_m x[b,n,m,:])
//           + per-utterance squared norms xnorm2[b, n*M+m] = ||x[b,n,m,:]||^2
// One block of 128 threads per (b,n); each thread owns 4 of the 512 dims.
// ---------------------------------------------------------------------------
__global__ void __launch_bounds__(128) ge2e_centroid_kernel(
    const float* __restrict__ x, float* __restrict__ c, float* __restrict__ xnorm2) {
  const int bn = blockIdx.x;                 // b*N_ + n
  const int t  = threadIdx.x;                // 0..127
  const float* base = x + (size_t)bn * M_ * D_;
  __shared__ float red[128];

  float a0 = 0.f, a1 = 0.f, a2 = 0.f, a3 = 0.f;
  for (int m = 0; m < M_; ++m) {
    const float* row = base + m * D_;
    float e0 = row[t], e1 = row[t + 128], e2 = row[t + 256], e3 = row[t + 384];
    a0 += e0; a1 += e1; a2 += e2; a3 += e3;
    red[t] = e0*e0 + e1*e1 + e2*e2 + e3*e3;
    __syncthreads();
    for (int off = 64; off > 0; off >>= 1) {
      if (t < off) red[t] += red[t + off];
      __syncthreads();
    }
    if (t == 0) xnorm2[(size_t)bn * M_ + m] = red[0];
    __syncthreads();
  }
  const float invM = 1.0f / (float)M_;
  a0 *= invM; a1 *= invM; a2 *= invM; a3 *= invM;
  red[t] = a0*a0 + a1*a1 + a2*a2 + a3*a3;
  __syncthreads();
  for (int off = 64; off > 0; off >>= 1) {
    if (t < off) red[t] += red[t + off];
    __syncthreads();
  }
  const float inv = 1.0f / fmaxf(sqrtf(red[0]), EPSF);
  float* crow = c + (size_t)bn * D_;
  crow[t]       = a0 * inv;
  crow[t + 128] = a1 * inv;
  crow[t + 256] = a2 * inv;
  crow[t + 384] = a3 * inv;
}

// ---------------------------------------------------------------------------
// Kernel 2: g[b,k,n] = x[b,k,:] . c[b,n,:]  (raw dot; row scaling deferred)
// One wave per 16x16 output tile, V_WMMA_F32_16X16X4_F32, K-loop of 128 steps.
// A 16x4 f32 layout: lanes 0-15 hold M=row, K={0,1} in v[0],v[1];
//                    lanes 16-31 hold same rows, K={2,3}.
// B 4x16 f32 layout (symmetric): lanes hold col=lane&15, K pair by lane>>4.
// D 16x16 f32 layout: VGPR v -> row v + 8*(lane>>4), col = lane&15.
// ---------------------------------------------------------------------------
__global__ void __launch_bounds__(256) ge2e_gemm_wmma(
    const float* __restrict__ x, const float* __restrict__ c,
    float* __restrict__ g) {
  const int lane  = threadIdx.x & 31;
  const int gwave = blockIdx.x * (blockDim.x >> 5) + (threadIdx.x >> 5);
  // gwave = b*160 + mtile*4 + ntile   (mtile: 40 row-tiles, ntile: 4 col-tiles)
  const int ntile = gwave & 3;
  const int mtile = (gwave >> 2) % 40;
  const int b     = gwave / 160;
  if (b >= B_) return;

  const int row   = lane & 15;             // tile-local row (A) / col (B)
  const int khalf = (lane >> 4) << 1;      // 0 or 2

  const float* xrow = x + ((size_t)b * NM_ + mtile * 16 + row) * D_;
  const float* crow = c + ((size_t)b * N_  + ntile * 16 + row) * D_;

  v8f acc = {0.f, 0.f, 0.f, 0.f, 0.f, 0.f, 0.f, 0.f};
  #pragma unroll 4
  for (int k0 = 0; k0 < D_; k0 += 4) {
    v2f av = *(const v2f*)(xrow + k0 + khalf);
    v2f bv = *(const v2f*)(crow + k0 + khalf);
    acc = __builtin_amdgcn_wmma_f32_16x16x4_f32(
        /*neg_a=*/false, av, /*neg_b=*/false, bv,
        /*c_mod=*/(short)0, acc, /*reuse_a=*/false, /*reuse_b=*/false);
  }

  const int colg  = ntile * 16 + (lane & 15);
  const int rbase = mtile * 16 + ((lane >> 4) << 3);
  float* gout = g + (size_t)b * NM_ * N_;
  #pragma unroll
  for (int v = 0; v < 8; ++v) {
    gout[(size_t)(rbase + v) * N_ + colg] = acc[v];
  }
}

// ---------------------------------------------------------------------------
// Kernel 3: zero the scalar accumulator
// ---------------------------------------------------------------------------
__global__ void ge2e_zero_kernel(float* __restrict__ out) {
  if (threadIdx.x == 0 && blockIdx.x == 0) out[0] = 0.0f;
}

// ---------------------------------------------------------------------------
// Kernel 4: per-row log-softmax cross-entropy with leave-one-out diagonal.
// One wave per row (b,k); 64 columns -> 2 per lane; wave shuffles for
// max / sum-exp reductions; lane 0 atomically adds the mean contribution.
// ---------------------------------------------------------------------------
__global__ void __launch_bounds__(256) ge2e_loss_kernel(
    const float* __restrict__ g, const float* __restrict__ xnorm2,
    const float* __restrict__ wp, const float* __restrict__ bp,
    float* __restrict__ out) {
  const int lane = threadIdx.x & 31;
  const int row  = blockIdx.x * (blockDim.x >> 5) + (threadIdx.x >> 5);
  if (row >= B_ * NM_) return;

  const int k   = row % NM_;
  const int spk = k / M_;               // speaker_ids = repeat(arange(N), M)
  const float w  = wp[0];
  const float bb = bp[0];

  const float* grow = g + (size_t)row * N_;
  const float nx2  = xnorm2[row];
  const float nx   = sqrtf(nx2);
  const float rinv = 1.0f / fmaxf(nx, EPSF);

  // leave-one-out: u = (M*c - x)/(M-1); excl = (x/||x||) . (u/max(||u||,eps))
  const float draw = grow[spk];                       // x . c_spk (raw)
  const float num  = ((float)M_ * draw - nx2) / (float)(M_ - 1);          // x.u
  const float uden = sqrtf((float)(M_ * M_) - 2.0f * (float)M_ * draw + nx2)
                     / (float)(M_ - 1);                                   // ||u||
  const float excl = (num / fmaxf(uden, EPSF)) * rinv;

  const int c0 = lane, c1 = lane + 32;
  const float s0 = (c0 == spk) ? excl : grow[c0] * rinv;
  const float s1 = (c1 == spk) ? excl : grow[c1] * rinv;
  const float l0 = w * s0 + bb;
  const float l1 = w * s1 + bb;

  float mx = fmaxf(l0, l1);
  #pragma unroll
  for (int off = 16; off > 0; off >>= 1) mx = fmaxf(mx, __shfl_xor(mx, off));
  float se = expf(l0 - mx) + expf(l1 - mx);
  #pragma unroll
  for (int off = 16; off > 0; off >>= 1) se += __shfl_xor(se, off);

  const float lse = mx + logf(se);
  const float ltgt = w * excl + bb;     // target column holds the excl value
  if (lane == 0) {
    atomicAdd(out, (lse - ltgt) * (1.0f / (float)(B_ * NM_)));
  }
}

// ---------------------------------------------------------------------------
extern "C" void kernel_launch(void* const* d_in, const int* in_sizes, int n_in,
                              void* d_out, int out_size, void* d_ws, size_t ws_size,
                              hipStream_t stream) {
  (void)in_sizes; (void)n_in; (void)out_size; (void)ws_size;
  const float* x  = (const float*)d_in[0];
  const float* w  = (const float*)d_in[1];
  const float* bb = (const float*)d_in[2];
  float* out = (float*)d_out;

  float* ws  = (float*)d_ws;
  float* c   = ws;                                   // B*N*D   = 2,097,152 f
  float* xn2 = c   + (size_t)B_ * N_ * D_;           // B*NM    =    40,960 f
  float* g   = xn2 + (size_t)B_ * NM_;               // B*NM*N  = 2,621,440 f

  // 1) centroids + row norms: one block per (b, n)
  ge2e_centroid_kernel<<<B_ * N_, 128, 0, stream>>>(x, c, xn2);

  // 2) WMMA GEMM: 64 batches * 40 * 4 tiles = 10240 waves, 8 waves/block
  ge2e_gemm_wmma<<<(B_ * 40 * 4) / 8, 256, 0, stream>>>(x, c, g);

  // 3) zero scalar, 4) per-row CE + mean
  ge2e_zero_kernel<<<1, 32, 0, stream>>>(out);
  ge2e_loss_kernel<<<(B_ * NM_) / 8, 256, 0, stream>>>(g, xn2, w, bb, out);
}